// ContrastiveAttention_77171972374780
// MI455X (gfx1250) — compile-verified
//
#include <hip/hip_runtime.h>
#include <math.h>

// ---------------------------------------------------------------------------
// ContrastiveAttention for MI455X (gfx1250, wave32)
// B=32, NI=1024, D=1024, P=1000, R=3   (all f32)
//
// Memory-bound stage: att_feats read (128MB) + out write (128MB) -> NT b128.
// Compute stages: f32 WMMA (V_WMMA_F32_16X16X4_F32) tile GEMMs, M in {32,96}.
// Each wave computes a 16x64 C slab: 1 A fragment -> 4 independent WMMAs.
// ---------------------------------------------------------------------------

typedef __attribute__((ext_vector_type(2))) float v2f;
typedef __attribute__((ext_vector_type(4))) float f32x4;
typedef __attribute__((ext_vector_type(8))) float v8f;

#define Bc   32
#define NIc  1024
#define Dc   1024
#define Pc   1000
#define Rc   3

// ---------------------------------------------------------------------------
// Templated wave-per-16x64-slab WMMA f32 GEMM:
//   C[m,n] = act( alpha * sum_k A[m,k] * Bmat[k,n] + bias[n] )
//   BTRANS   : Bmat[k,n] = B[n*ldb + k]          (X @ W^T, row-major W[N,K])
//   USE_KMOD : Bmat[k,n] = B[(k % kmod)*ldb + n] (pool repeated along K)
// M multiple of 16; K multiple of 4; N ragged allowed (clamped addressing:
// garbage only reaches discarded C columns, EXEC stays all-ones for WMMA).
// blockDim.x = 128 (4 waves).
// ---------------------------------------------------------------------------
template <bool BTRANS, bool USE_KMOD, bool HASBIAS, bool RELU>
__global__ __launch_bounds__(128)
void gemm_wmma(const float* __restrict__ A, int lda, long long sA,
               const float* __restrict__ B, int ldb, long long sB,
               float* __restrict__ C, int ldc, long long sC,
               int M, int N, int K, int kmod,
               float alpha, const float* __restrict__ bias)
{
    const int mtiles  = M >> 4;
    const int ngroups = (N + 63) >> 6;                  // 4 x 16-col tiles per wave
    const int wid = threadIdx.x >> 5;
    const int g   = blockIdx.x * (blockDim.x >> 5) + wid;
    if (g >= mtiles * ngroups) return;                  // uniform per wave

    const float* Ab = A + (size_t)blockIdx.z * (size_t)sA;
    const float* Bb = B + (size_t)blockIdx.z * (size_t)sB;
    float*       Cb = C + (size_t)blockIdx.z * (size_t)sC;

    const int mt   = g / ngroups;
    const int ng   = g % ngroups;
    const int lane = threadIdx.x & 31;
    const int half = lane >> 4;                         // 0: K {k,k+1}, 1: {k+2,k+3}
    const int l    = lane & 15;
    const int row  = (mt << 4) + l;                     // A row for this lane
    const int col0 = (ng << 6) + l;                     // C col of sub-tile 0

    // Clamped columns for addressing (safe loads; junk lands in dropped cols)
    int colc[4];
#pragma unroll
    for (int j = 0; j < 4; ++j) {
        int c = col0 + 16 * j;
        colc[j] = (c < N) ? c : (N - 1);
    }

    v8f acc[4];
    const v8f z = {0.f, 0.f, 0.f, 0.f, 0.f, 0.f, 0.f, 0.f};
#pragma unroll
    for (int j = 0; j < 4; ++j) acc[j] = z;

    const float* ap = Ab + (size_t)row * lda + (half << 1);

    if (BTRANS) {
        const float* bp0 = Bb + (size_t)colc[0] * ldb + (half << 1);
        const float* bp1 = Bb + (size_t)colc[1] * ldb + (half << 1);
        const float* bp2 = Bb + (size_t)colc[2] * ldb + (half << 1);
        const float* bp3 = Bb + (size_t)colc[3] * ldb + (half << 1);
#pragma unroll 2
        for (int k0 = 0; k0 < K; k0 += 4) {
            v2f a  = *(const v2f*)ap;  ap  += 4;
            v2f b0 = *(const v2f*)bp0; bp0 += 4;
            v2f b1 = *(const v2f*)bp1; bp1 += 4;
            v2f b2 = *(const v2f*)bp2; bp2 += 4;
            v2f b3 = *(const v2f*)bp3; bp3 += 4;
            acc[0] = __builtin_amdgcn_wmma_f32_16x16x4_f32(false, a, false, b0, (short)0, acc[0], false, false);
            acc[1] = __builtin_amdgcn_wmma_f32_16x16x4_f32(false, a, false, b1, (short)0, acc[1], false, false);
            acc[2] = __builtin_amdgcn_wmma_f32_16x16x4_f32(false, a, false, b2, (short)0, acc[2], false, false);
            acc[3] = __builtin_amdgcn_wmma_f32_16x16x4_f32(false, a, false, b3, (short)0, acc[3], false, false);
        }
    } else {
        // B row index p = (k0 + 2*half) % kmod tracked incrementally (no div)
        int p = (half << 1);
        for (int k0 = 0; k0 < K; k0 += 4) {
            int p1 = p + 1;
            if (p1 >= kmod) p1 -= kmod;
            v2f a = *(const v2f*)ap; ap += 4;
            const float* br0 = Bb + (size_t)p  * ldb;
            const float* br1 = Bb + (size_t)p1 * ldb;
            v2f b0, b1, b2, b3;
            b0.x = br0[colc[0]]; b0.y = br1[colc[0]];
            b1.x = br0[colc[1]]; b1.y = br1[colc[1]];
            b2.x = br0[colc[2]]; b2.y = br1[colc[2]];
            b3.x = br0[colc[3]]; b3.y = br1[colc[3]];
            acc[0] = __builtin_amdgcn_wmma_f32_16x16x4_f32(false, a, false, b0, (short)0, acc[0], false, false);
            acc[1] = __builtin_amdgcn_wmma_f32_16x16x4_f32(false, a, false, b1, (short)0, acc[1], false, false);
            acc[2] = __builtin_amdgcn_wmma_f32_16x16x4_f32(false, a, false, b2, (short)0, acc[2], false, false);
            acc[3] = __builtin_amdgcn_wmma_f32_16x16x4_f32(false, a, false, b3, (short)0, acc[3], false, false);
            p += 4;
            if (p >= kmod) p -= kmod;
        }
    }

    // Epilogue: C layout per tile: VGPR i -> row (i + 8*half), col = lane&15
#pragma unroll
    for (int j = 0; j < 4; ++j) {
        const int col = col0 + 16 * j;
        if (col < N) {
            const float bv = HASBIAS ? bias[col] : 0.f;
#pragma unroll
            for (int i = 0; i < 8; ++i) {
                const int r = (mt << 4) + i + (half << 3);
                float val = acc[j][i] * alpha + bv;
                if (RELU) val = fmaxf(val, 0.f);
                Cb[(size_t)r * ldc + col] = val;
            }
        }
    }
}

// ---------------------------------------------------------------------------
// Row softmax over P=1000 (one wave per (r,b) row), writing the result
// rearranged to w2[b, r*P + p] so v_agg becomes a plain row-major GEMM.
// ---------------------------------------------------------------------------
__global__ void softmax_rearrange(const float* __restrict__ scores,
                                  float* __restrict__ w2)
{
    const int rb = blockIdx.x;            // 0 .. R*B-1
    const int r  = rb / Bc;
    const int b  = rb % Bc;
    const int t  = threadIdx.x;           // 32 threads = 1 wave

    const float* srow = scores + (size_t)rb * Pc;

    float mx = -3.402823466e38f;
    for (int p = t; p < Pc; p += 32) mx = fmaxf(mx, srow[p]);
    for (int off = 16; off; off >>= 1) mx = fmaxf(mx, __shfl_xor(mx, off));

    float sum = 0.f;
    for (int p = t; p < Pc; p += 32) sum += expf(srow[p] - mx);
    for (int off = 16; off; off >>= 1) sum += __shfl_xor(sum, off);

    const float inv = 1.f / sum;
    float* wrow = w2 + (size_t)b * (Rc * Pc) + (size_t)r * Pc;
    for (int p = t; p < Pc; p += 32) wrow[p] = expf(srow[p] - mx) * inv;
}

// ---------------------------------------------------------------------------
__device__ inline float block_reduce_sum(float val)
{
    __shared__ float red[8];
    const int lane = threadIdx.x & 31;
    const int wid  = threadIdx.x >> 5;
    for (int off = 16; off; off >>= 1) val += __shfl_xor(val, off);
    if (lane == 0) red[wid] = val;
    __syncthreads();
    const int nw = (blockDim.x + 31) >> 5;
    if (wid == 0) {
        float v = (lane < nw) ? red[lane] : 0.f;
        for (int off = 4; off; off >>= 1) v += __shfl_xor(v, off);
        if (lane == 0) red[0] = v;
    }
    __syncthreads();
    return red[0];
}

// ---------------------------------------------------------------------------
// sim[b] = scale * <q_b, k_b>;  a_c[b,:] = fc[b,:] - sigmoid(sim[b]) * v[b,:]
// ---------------------------------------------------------------------------
__global__ void sim_vcommon(const float* __restrict__ q, const float* __restrict__ k,
                            const float* __restrict__ v, const float* __restrict__ fc,
                            float* __restrict__ a_c)
{
    const int b = blockIdx.x;
    const int t = threadIdx.x;            // 256
    const float* qr = q + (size_t)b * Dc;
    const float* kr = k + (size_t)b * Dc;

    float s = 0.f;
    for (int d = t; d < Dc; d += 256) s += qr[d] * kr[d];
    s = block_reduce_sum(s);

    __shared__ float sig_s;
    if (t == 0) sig_s = 1.f / (1.f + expf(-s * 0.03125f));   // scale = 1/sqrt(1024)
    __syncthreads();
    const float sig = sig_s;

    const float* fr = fc + (size_t)b * Dc;
    const float* vr = v  + (size_t)b * Dc;
    float*       ar = a_c + (size_t)b * Dc;
    for (int d = t; d < Dc; d += 256) ar[d] = fr[d] - sig * vr[d];
}

// ---------------------------------------------------------------------------
// sbuf[b] = <v_diff[b,:], Wg2>   (Wg2 = Wg + D)
// ---------------------------------------------------------------------------
__global__ void vdiff_dot_wg2(const float* __restrict__ v_diff,
                              const float* __restrict__ Wg,
                              float* __restrict__ sbuf)
{
    const int b = blockIdx.x;
    const int t = threadIdx.x;            // 256
    const float* vr = v_diff + (size_t)b * Dc;
    const float* w2 = Wg + Dc;
    float s = 0.f;
    for (int d = t; d < Dc; d += 256) s += vr[d] * w2[d];
    s = block_reduce_sum(s);
    if (t == 0) sbuf[b] = s;
}

// ---------------------------------------------------------------------------
// Bandwidth stage. One block per (b,n) row (32768 blocks x 256 thr).
// Each thread owns one float4 of att (NT b128 load), row dot with Wg1,
// block reduce, sigmoid gate, fused output (NT b128 store).
// ---------------------------------------------------------------------------
__global__ void fuse_gate_out(const float* __restrict__ att,
                              const float* __restrict__ Wg,
                              const float* __restrict__ sbuf,
                              const float* __restrict__ v_diff,
                              const float* __restrict__ bgp,
                              const float* __restrict__ rsp,
                              float* __restrict__ out)
{
    const int row = blockIdx.x;           // b*NI + n
    const int b   = row >> 10;            // NI = 1024
    const int t   = threadIdx.x;          // 256, 4 floats each

    const f32x4* arow = (const f32x4*)(att + (size_t)row * Dc);
    f32x4 a = __builtin_nontemporal_load(arow + t);       // stream att (read-once)
    f32x4 w = ((const f32x4*)Wg)[t];                      // Wg1 stays cached

    float dot = a.x * w.x + a.y * w.y + a.z * w.z + a.w * w.w;
    const float total = block_reduce_sum(dot);

    const float g = 1.f / (1.f + expf(-(total + sbuf[b] + bgp[0])));
    const float c = rsp[0] * g;

    f32x4 vd = ((const f32x4*)(v_diff + (size_t)b * Dc))[t];  // row cached, reused NI times
    f32x4 o;
    o.x = a.x + c * vd.x;
    o.y = a.y + c * vd.y;
    o.z = a.z + c * vd.z;
    o.w = a.w + c * vd.w;
    __builtin_nontemporal_store(o, (f32x4*)(out + (size_t)row * Dc) + t);  // stream out
}

// ---------------------------------------------------------------------------
static inline int gemm_blocks(int M, int N)
{
    const int waves = (M >> 4) * ((N + 63) >> 6);
    return (waves + 3) / 4;
}

extern "C" void kernel_launch(void* const* d_in, const int* in_sizes, int n_in,
                              void* d_out, int out_size, void* d_ws, size_t ws_size,
                              hipStream_t stream)
{
    const float* att   = (const float*)d_in[0];   // (B,NI,D)
    const float* fc    = (const float*)d_in[1];   // (B,D)
    const float* pool  = (const float*)d_in[2];   // (P,D)
    const float* Wagg  = (const float*)d_in[3];   // (R,D,D)
    const float* Wq    = (const float*)d_in[4];
    const float* bq    = (const float*)d_in[5];
    const float* Wk    = (const float*)d_in[6];
    const float* bk    = (const float*)d_in[7];
    const float* Wv    = (const float*)d_in[8];
    const float* bv    = (const float*)d_in[9];
    const float* Wc    = (const float*)d_in[10];
    const float* bc    = (const float*)d_in[11];
    const float* Wg    = (const float*)d_in[12];  // (2D,)
    const float* bg    = (const float*)d_in[13];  // scalar
    const float* rs    = (const float*)d_in[14];  // scalar
    float* out = (float*)d_out;

    // workspace layout (floats)
    float* ws     = (float*)d_ws;
    float* q_all  = ws;                       // R*B*D   = 98304
    float* scores = q_all  + Rc * Bc * Dc;    // R*B*P   = 96000
    float* w2     = scores + Rc * Bc * Pc;    // B*R*P   = 96000
    float* v_agg  = w2     + Bc * Rc * Pc;    // B*D
    float* qb     = v_agg  + Bc * Dc;
    float* kb     = qb     + Bc * Dc;
    float* vb     = kb     + Bc * Dc;
    float* a_c    = vb     + Bc * Dc;
    float* v_diff = a_c    + Bc * Dc;
    float* sbuf   = v_diff + Bc * Dc;         // B

    const float scale = 0.03125f;             // 1/sqrt(1024)

    // 1) q_all[r] = fc @ W_agg[r]^T        (batched over r via grid.z)
    gemm_wmma<true, false, false, false><<<dim3(gemm_blocks(Bc, Dc), 1, Rc), 128, 0, stream>>>(
        fc, Dc, 0LL, Wagg, Dc, (long long)Dc * Dc,
        q_all, Dc, (long long)Bc * Dc,
        Bc, Dc, Dc, 0, 1.0f, nullptr);

    // 2) scores = (q_all as (R*B) x D) @ pool^T * scale
    gemm_wmma<true, false, false, false><<<dim3(gemm_blocks(Rc * Bc, Pc)), 128, 0, stream>>>(
        q_all, Dc, 0LL, pool, Dc, 0LL,
        scores, Pc, 0LL,
        Rc * Bc, Pc, Dc, 0, scale, nullptr);

    // 3) softmax rows -> w2[b, r*P+p]
    softmax_rearrange<<<Rc * Bc, 32, 0, stream>>>(scores, w2);

    // 4) v_agg = (1/R) * w2 (B x R*P) @ pool-repeated (R*P x D)   (kmod=P)
    gemm_wmma<false, true, false, false><<<dim3(gemm_blocks(Bc, Dc)), 128, 0, stream>>>(
        w2, Rc * Pc, 0LL, pool, Dc, 0LL,
        v_agg, Dc, 0LL,
        Bc, Dc, Rc * Pc, Pc, 1.0f / (float)Rc, nullptr);

    // 5) q = fc @ Wq^T + bq ; k = v_agg @ Wk^T + bk ; v = v_agg @ Wv^T + bv
    gemm_wmma<true, false, true, false><<<dim3(gemm_blocks(Bc, Dc)), 128, 0, stream>>>(
        fc, Dc, 0LL, Wq, Dc, 0LL, qb, Dc, 0LL, Bc, Dc, Dc, 0, 1.0f, bq);
    gemm_wmma<true, false, true, false><<<dim3(gemm_blocks(Bc, Dc)), 128, 0, stream>>>(
        v_agg, Dc, 0LL, Wk, Dc, 0LL, kb, Dc, 0LL, Bc, Dc, Dc, 0, 1.0f, bk);
    gemm_wmma<true, false, true, false><<<dim3(gemm_blocks(Bc, Dc)), 128, 0, stream>>>(
        v_agg, Dc, 0LL, Wv, Dc, 0LL, vb, Dc, 0LL, Bc, Dc, Dc, 0, 1.0f, bv);

    // 6) a_c = fc - sigmoid(scale * <q,k>) * v
    sim_vcommon<<<Bc, 256, 0, stream>>>(qb, kb, vb, fc, a_c);

    // 7) v_diff = relu(a_c @ Wc^T + bc)
    gemm_wmma<true, false, true, true><<<dim3(gemm_blocks(Bc, Dc)), 128, 0, stream>>>(
        a_c, Dc, 0LL, Wc, Dc, 0LL, v_diff, Dc, 0LL, Bc, Dc, Dc, 0, 1.0f, bc);

    // 8) sbuf[b] = <v_diff[b,:], Wg2>
    vdiff_dot_wg2<<<Bc, 256, 0, stream>>>(v_diff, Wg, sbuf);

    // 9) out = att + rs * sigmoid(att@Wg1 + sbuf[b] + bg) * v_diff   (streaming)
    fuse_gate_out<<<Bc * NIc, 256, 0, stream>>>(att, Wg, sbuf, v_diff, bg, rs, out);
}